// Cluster_72292889527017
// MI455X (gfx1250) — compile-verified
//
#include <hip/hip_runtime.h>

typedef float v2f __attribute__((ext_vector_type(2)));
typedef float v8f __attribute__((ext_vector_type(8)));

#define NROWS   65536
#define KC      256     // number of centroids (output columns)
#define DD      256     // feature dim
#define KA      260     // augmented K: [z | z2 | 1 | 0 | 0]
#define KTILES  65      // KA / 4
#define QBLOCKS 512     // main-kernel grid
#define STRIPS_PER_WG 8 // 512 blocks * 8 strips * 16 rows = 65536 rows

// ---------------- prep: build augmented, transposed centroid matrix ----------------
// ct[k][j], k in [0,260): rows 0..255 = -2*centroids[j][k]; row 256 = 1.0
// (pairs with z2 slot); row 257 = c2[j] (pairs with the 1.0 slot); 258/259 = 0.
__global__ __launch_bounds__(256)
void prep_ct_kernel(const float* __restrict__ cen, float* __restrict__ ct) {
  const int j = threadIdx.x;
  const int d = blockIdx.x;
  if (d < DD) {
    ct[(size_t)d * KC + j] = -2.0f * cen[(size_t)j * DD + d];
  } else {
    ct[(size_t)256 * KC + j] = 1.0f;
  }
}

__global__ __launch_bounds__(256)
void prep_c2_kernel(const float* __restrict__ cen, float* __restrict__ ct) {
  const int j = threadIdx.x;
  float s = 0.0f;
  for (int d = 0; d < DD; ++d) {
    const float v = cen[(size_t)j * DD + d];
    s = fmaf(v, v, s);
  }
  ct[(size_t)257 * KC + j] = s;
  ct[(size_t)258 * KC + j] = 0.0f;
  ct[(size_t)259 * KC + j] = 0.0f;
}

// Issue async global->LDS copy of one 16x256 strip into LDS buffer (augmented
// row stride KA). Each of the 512 threads issues two B128 transfers.
// LDS byte offsets r*1040 + 4*c are 16B-aligned (1040 = 65*16).
__device__ __forceinline__ void issue_strip_async(const float* __restrict__ z,
                                                  float* ldsbuf, int r0, int tid) {
#pragma unroll
  for (int i = 0; i < 2; ++i) {
    const int idx = tid + (i << 9);          // 0..1023 chunks of 4 floats
    const int r   = idx >> 6;                // row 0..15
    const int c   = (idx & 63) << 2;         // float column, multiple of 4
    const unsigned lofs = (unsigned)(size_t)(ldsbuf + r * KA + c);
    const float* g = z + (size_t)(r0 + r) * DD + c;
    asm volatile("global_load_async_to_lds_b128 %0, %1, off"
                 :: "v"(lofs), "v"(g) : "memory");
  }
}

// ---------------- main: fused GEMM-distance + q + row-normalize ----------------
__global__ __launch_bounds__(512, 1)
void q_kernel(const float* __restrict__ z, const float* __restrict__ ct,
              float* __restrict__ qout, float* __restrict__ cpart) {
  __shared__ __attribute__((aligned(16))) float sA[2][16 * KA]; // double buffer
  __shared__ float sPart[16 * 16]; // per-wave row-sum partials
  __shared__ float sRow[16];       // combined row sums

  const int tid  = threadIdx.x;
  const int wave = tid >> 5;
  const int lane = tid & 31;
  const int half = lane >> 4;      // which half-wave (K split of WMMA layout)
  const int m    = lane & 15;
  const int c0   = wave << 4;      // this wave's column-tile base

  // B fragments for this wave's 16 columns, all 65 K-steps, register-resident.
  // Layout per ISA: VGPR0 = {K=k | K=k+2}, VGPR1 = {K=k+1 | K=k+3}, lanes = N.
  v2f bfrag[KTILES];
#pragma unroll
  for (int t = 0; t < KTILES; ++t) {
    const int k = t * 4 + 2 * half;
    bfrag[t].x = ct[(size_t)k * KC + c0 + m];
    bfrag[t].y = ct[(size_t)(k + 1) * KC + c0 + m];
  }

  const int strip0 = blockIdx.x * STRIPS_PER_WG;
  float csum = 0.0f;  // per-lane column-sum accumulator (column c0+m)

  // prologue: start strip 0 transfer
  issue_strip_async(z, sA[0], strip0 << 4, tid);

  for (int s = 0; s < STRIPS_PER_WG; ++s) {
    const int cur = s & 1;
    const int r0  = (strip0 + s) << 4;
    float* A = sA[cur];

    // wait for this wave's async transfers, then cross-wave barrier
    asm volatile("s_wait_asynccnt 0" ::: "memory");
    __syncthreads();

    // wave w computes z2 of row w; fill augmented columns
    {
      float zp = 0.0f;
#pragma unroll
      for (int i = 0; i < 8; ++i) {
        const float v = A[wave * KA + lane + (i << 5)];
        zp = fmaf(v, v, zp);
      }
#pragma unroll
      for (int off = 16; off >= 1; off >>= 1) zp += __shfl_xor(zp, off, 32);
      if (lane == 0) {
        A[wave * KA + 256] = zp;
        A[wave * KA + 257] = 1.0f;
        A[wave * KA + 258] = 0.0f;
        A[wave * KA + 259] = 0.0f;
      }
    }
    __syncthreads();

    // prefetch next strip into the other buffer while we compute on this one
    if (s + 1 < STRIPS_PER_WG)
      issue_strip_async(z, sA[cur ^ 1], (strip0 + s + 1) << 4, tid);

    // sse tile = augmented A x B via chained fp32 WMMA (K = 260)
    v8f acc = {0.f, 0.f, 0.f, 0.f, 0.f, 0.f, 0.f, 0.f};
#pragma unroll
    for (int t = 0; t < KTILES; ++t) {
      const int k = t * 4 + 2 * half;
      v2f a;
      a.x = A[m * KA + k];        // A layout: lanes = M, VGPR0={K=k|K=k+2}
      a.y = A[m * KA + k + 1];
      acc = __builtin_amdgcn_wmma_f32_16x16x4_f32(
          false, a, false, bfrag[t], (short)0, acc, false, false);
    }

    // q_un = 1/(1+max(sse,0))
    float qv[8];
#pragma unroll
    for (int j = 0; j < 8; ++j) {
      const float sse = fmaxf(acc[j], 0.0f);
      qv[j] = 1.0f / (1.0f + sse);
    }

    // partial row sums over this wave's 16 columns (reduce within 16-lane halves)
    float rs[8];
#pragma unroll
    for (int j = 0; j < 8; ++j) {
      float r = qv[j];
      r += __shfl_xor(r, 1, 32);
      r += __shfl_xor(r, 2, 32);
      r += __shfl_xor(r, 4, 32);
      r += __shfl_xor(r, 8, 32);
      rs[j] = r;  // lanes 0-15: row j; lanes 16-31: row j+8
    }
    if (m == 0) {
#pragma unroll
      for (int j = 0; j < 8; ++j) sPart[wave * 16 + (half << 3) + j] = rs[j];
    }
    __syncthreads();

    // deterministic combine across the 16 column tiles
    if (tid < 16) {
      float tsum = 0.0f;
#pragma unroll
      for (int w = 0; w < 16; ++w) tsum += sPart[w * 16 + tid];
      sRow[tid] = tsum;
    }
    __syncthreads();

    // normalize, store q (keep in L2: re-read by p_kernel), accumulate partials
#pragma unroll
    for (int j = 0; j < 8; ++j) {
      const int row = (half << 3) + j;
      const float qn = qv[j] / sRow[row];
      qout[(size_t)(r0 + row) * KC + c0 + m] = qn;
      csum += qn;
    }
    // no trailing barrier needed: next strip's first barrier orders LDS reuse
  }

  // fold the two half-wave lanes holding the same column, one store per column
  csum += __shfl_xor(csum, 16, 32);
  if (half == 0) cpart[(size_t)blockIdx.x * KC + c0 + m] = csum;
}

// ---------------- fixed-order column-sum reduction (deterministic) ----------------
__global__ __launch_bounds__(256)
void colreduce_kernel(const float* __restrict__ cpart, float* __restrict__ colsum) {
  const int j = threadIdx.x;
  float s = 0.0f;
  for (int b = 0; b < QBLOCKS; ++b) s += cpart[(size_t)b * KC + j];
  colsum[j] = s;
}

// ---------------- p = rownorm(q^2 / colsum) ----------------
__global__ __launch_bounds__(256)
void p_kernel(const float* __restrict__ q, const float* __restrict__ colsum,
              float* __restrict__ p) {
  __shared__ float sInv[KC];
  const int tid = threadIdx.x;
  sInv[tid] = 1.0f / colsum[tid];
  __syncthreads();
  const int wave = tid >> 5;
  const int lane = tid & 31;
  const size_t row = (size_t)blockIdx.x * 8 + wave;
  float sf[8];
  float tot = 0.0f;
#pragma unroll
  for (int i = 0; i < 8; ++i) {
    const int c = lane + (i << 5);
    const float qv = q[row * KC + c];
    sf[i] = qv * qv * sInv[c];
    tot += sf[i];
  }
#pragma unroll
  for (int off = 16; off >= 1; off >>= 1) tot += __shfl_xor(tot, off, 32);
  const float inv = 1.0f / tot;
  // p is write-once, never re-read: nontemporal store preserves L2 for q
#pragma unroll
  for (int i = 0; i < 8; ++i)
    __builtin_nontemporal_store(sf[i] * inv, &p[row * KC + lane + (i << 5)]);
}

extern "C" void kernel_launch(void* const* d_in, const int* in_sizes, int n_in,
                              void* d_out, int out_size, void* d_ws, size_t ws_size,
                              hipStream_t stream) {
  const float* z   = (const float*)d_in[0];  // (65536, 256) fp32
  const float* cen = (const float*)d_in[1];  // (256, 256) fp32

  float* p = (float*)d_out;                  // outputs: (p, q) concatenated
  float* q = p + (size_t)NROWS * KC;

  float* ct     = (float*)d_ws;              // 260*256 floats
  float* colsum = ct + (size_t)KA * KC;      // 256 floats
  float* cpart  = colsum + KC;               // QBLOCKS*256 floats

  prep_ct_kernel<<<DD + 1, 256, 0, stream>>>(cen, ct);
  prep_c2_kernel<<<1, 256, 0, stream>>>(cen, ct);
  q_kernel<<<QBLOCKS, 512, 0, stream>>>(z, ct, q, cpart);
  colreduce_kernel<<<1, 256, 0, stream>>>(cpart, colsum);
  p_kernel<<<NROWS / 8, 256, 0, stream>>>(q, colsum, p);
}